// DCF_89653147336836
// MI455X (gfx1250) — compile-verified
//
#include <hip/hip_runtime.h>
#include <hip/hip_bf16.h>
#include <math.h>

#define NB    8     // NUM_BLOCK
#define NT    500   // NUM_FRAME
#define NF    257   // NUM_BIN
#define NC    8     // NUM_CH
#define NN    4     // NUM_NULL
#define LOWF  5
#define HIGHF 70
#define ALPHA 0.35f

#define AS1 __attribute__((address_space(1)))
#define AS3 __attribute__((address_space(3)))

typedef __attribute__((ext_vector_type(2))) float v2f;
typedef __attribute__((ext_vector_type(8))) float v8f;
typedef int v4i __attribute__((vector_size(4 * sizeof(int))));

#if defined(__HIP_DEVICE_COMPILE__) && __has_builtin(__builtin_amdgcn_global_load_async_to_lds_b128)
#define HAVE_ASYNC_LDS 1
#else
#define HAVE_ASYNC_LDS 0
#endif

#if HAVE_ASYNC_LDS
#if __has_builtin(__builtin_amdgcn_s_wait_asynccnt)
#define WAIT_ASYNC() __builtin_amdgcn_s_wait_asynccnt(0)
#else
#define WAIT_ASYNC() asm volatile("s_wait_asynccnt 0x0" ::: "memory")
#endif
#else
#define WAIT_ASYNC() ((void)0)
#endif

__device__ __forceinline__ float clip01(float v) {
    return fminf(fmaxf(v, 0.01f), 1.0f);
}

// ---------------------------------------------------------------------------
// Stage 1: one workgroup per block b. Threads = (f - LOWF)*4 + n for the 65
// bins that feed the ratio reduction. Each thread owns one (f,n) IIR chain;
// per time step an LDS ds_add_f32 reduction forms pre/aft and ratio[b][t][n].
// ---------------------------------------------------------------------------
__global__ __launch_bounds__(288)
void dcf_stage1_ratio(const float* __restrict__ x,
                      const int*   __restrict__ beam_id,
                      const float* __restrict__ targ_w,
                      const float* __restrict__ null_w,
                      float*       __restrict__ ratio_ws) {
    const int b    = blockIdx.x;
    const int beam = beam_id[b];
    const int tid  = threadIdx.x;
    const int fl   = tid >> 2;          // 0..64 (65 bins)
    const int n    = tid & 3;
    const bool act = (fl < (HIGHF - LOWF));
    const int f    = LOWF + (act ? fl : 0);

    __shared__ float s_aft[NN];
    __shared__ float s_pre;

    float twr[NC], twi[NC], nwr[NC], nwi[NC];
    if (act) {
#pragma unroll
        for (int c = 0; c < NC; ++c) {
            twr[c] = targ_w[(((size_t)beam*2 + 0)*NF + f)*NC + c];
            twi[c] = targ_w[(((size_t)beam*2 + 1)*NF + f)*NC + c];
            nwr[c] = null_w[((((size_t)beam*NN + n)*2 + 0)*NF + f)*NC + c];
            nwi[c] = null_w[((((size_t)beam*NN + n)*2 + 1)*NF + f)*NC + c];
        }
    }

    float phi_r = 0.f, phi_i = 0.f, psd = 0.f;
    const float a = ALPHA, oma = 1.0f - ALPHA;

    for (int t = 0; t < NT; ++t) {
        if (tid < NN) s_aft[tid] = 0.f;
        if (tid == NN) s_pre = 0.f;
        __syncthreads();

        if (act) {
            const float* xr = x + ((((size_t)b*NT + t)*2 + 0)*NF + f)*NC;
            const float* xi = x + ((((size_t)b*NT + t)*2 + 1)*NF + f)*NC;
            float tr = 0.f, ti = 0.f, nr = 0.f, ni = 0.f, pw = 0.f;
#pragma unroll
            for (int c = 0; c < NC; ++c) {
                float ar = xr[c], ai = xi[c];
                tr += ar*twr[c] - ai*twi[c];
                ti += ar*twi[c] + ai*twr[c];
                nr += ar*nwr[c] - ai*nwi[c];
                ni += ar*nwi[c] + ai*nwr[c];
                pw += ar*ar + ai*ai;
            }
            pw *= (1.0f / NC);
            if (t == 0) {   // reference's asymmetric init (only first term scaled)
                phi_r = oma*tr*nr + ti*ni;
                phi_i = oma*ti*nr - tr*ni;
                psd   = oma*pw;
            } else {
                phi_r = a*phi_r + oma*(tr*nr + ti*ni);
                phi_i = a*phi_i + oma*(ti*nr - tr*ni);
                psd   = a*psd   + oma*pw;
            }
            float phi = sqrtf(phi_r*phi_r + phi_i*phi_i);
            atomicAdd(&s_aft[n], phi);
            if (n == 0) atomicAdd(&s_pre, psd);
        }
        __syncthreads();

        if (tid < NN) {
            float r = clip01(s_aft[tid] / (s_pre + 1e-10f));
            ratio_ws[((size_t)b*NT + t)*NN + tid] = r;   // t==0 written but unused
        }
    }
}

// Build B(k, j) for the 16x16 weight matrix of frequency f:
//   A columns k: k<8 -> xr[c=k], k>=8 -> xi[c=k-8]
//   B columns j: 0=tr, 1=ti, 2..5=nr[n], 6..9=ni[n], 10..15=zero pad
__device__ __forceinline__ float wmat(const float* __restrict__ targ_w,
                                      const float* __restrict__ null_w,
                                      int beam, int f, int k, int j) {
    if (j >= 10) return 0.f;
    const int ri = k >> 3, c = k & 7;
    if (j < 2) {
        float wr = targ_w[(((size_t)beam*2 + 0)*NF + f)*NC + c];
        float wi = targ_w[(((size_t)beam*2 + 1)*NF + f)*NC + c];
        if (j == 0) return ri == 0 ? wr : -wi;   // tr = xr.wr - xi.wi
        else        return ri == 0 ? wi :  wr;   // ti = xr.wi + xi.wr
    } else {
        const int  n    = (j - 2) & 3;
        const bool isNi = (j >= 6);
        float wr = null_w[((((size_t)beam*NN + n)*2 + 0)*NF + f)*NC + c];
        float wi = null_w[((((size_t)beam*NN + n)*2 + 1)*NF + f)*NC + c];
        if (!isNi) return ri == 0 ? wr : -wi;
        else       return ri == 0 ? wi :  wr;
    }
}

// ---------------------------------------------------------------------------
// Stage 2: one WAVE per (b, f). Per 16-frame tile:
//   - the x tile (2 ri x 16 frames x 8 bins x 8 ch = 8 KB) is staged into LDS
//     by coalesced async Global->LDS B128 copies, double-buffered so the DMA of
//     tile tt+1 overlaps the compute of tile tt (ASYNCcnt / s_wait_asynccnt).
//   - A = x[16 frames x 16 chRI] read from LDS (bank-rotated, conflict-free),
//     B = per-f weight matrix (built once), D = A*B via 4x
//     v_wmma_f32_16x16x4_f32 (K=16). D -> LDS, then lanes 0-3 run the
//     sequential 16-step recurrence for n=0..3 using the precomputed ratio.
// ---------------------------------------------------------------------------
#define XROW 68                       // 64 data dwords + 4 pad (16B align + bank rotate)

__global__ __launch_bounds__(256)
void dcf_stage2_main(const float* __restrict__ x,
                     const int*   __restrict__ beam_id,
                     const float* __restrict__ targ_w,
                     const float* __restrict__ null_w,
                     const float* __restrict__ ratio_ws,
                     float*       __restrict__ dcf_out,
                     float*       __restrict__ targ_out) {
    const int b     = blockIdx.x;
    const int beam  = beam_id[b];
    const int wave  = threadIdx.x >> 5;
    const int lane  = threadIdx.x & 31;
    const int j     = lane & 15;         // A row (M) / B,D column (N)
    const int hi    = lane >> 4;         // upper half-lane group
    const int fbase = blockIdx.y * 8;
    const int f     = fbase + wave;
    const bool vf   = (f < NF);
    const int fc    = vf ? f : (NF - 1); // clamp addressing for invalid waves

    __shared__ float s_x[2][2 * 16 * XROW];   // [buf][ri*16+tf][XROW] (2 x 8704 B)
    __shared__ float s_D[8][16][12];          // [wave][frame-in-tile][output col]
    __shared__ float s_pw[8][16];

    // Cooperative staging of one 16-frame tile: 32 rows x 256 B, 16 B per lane.
    auto stage_tile = [&](int tt, int bufsel) {
#pragma unroll
        for (int it = 0; it < 2; ++it) {
            const int q   = it * 256 + (int)threadIdx.x;   // 0..511
            const int row = q >> 4;                        // ri*16 + tf
            const int sub = q & 15;                        // 16B chunk in row
            const int ri  = row >> 4;
            const int tf  = row & 15;
            int t = tt * 16 + tf; t = (t < NT) ? t : (NT - 1);
            int foff = fbase * NC + sub * 4;               // element offset in (b,t,ri) row
            const int fmaxoff = NF * NC - 4;
            foff = (foff < fmaxoff) ? foff : fmaxoff;      // clamp f edge (garbage ok)
            const float* g = x + (((size_t)b*NT + t)*2 + ri)*(size_t)(NF*NC) + foff;
            float* l = &s_x[bufsel][row * XROW + sub * 4]; // 16B-aligned (XROW*4 = 272)
#if HAVE_ASYNC_LDS
            __builtin_amdgcn_global_load_async_to_lds_b128((AS1 v4i*)g,
                                                           (AS3 v4i*)l, 0, 0);
#else
            *reinterpret_cast<float4*>(l) = *reinterpret_cast<const float4*>(g);
#endif
        }
    };

    // Per-lane B fragments (reused across all 32 frame tiles)
    v2f Bf[4];
#pragma unroll
    for (int kb = 0; kb < 4; ++kb) {
        const int k0 = kb*4 + hi*2;
        Bf[kb].x = wmat(targ_w, null_w, beam, fc, k0,     j);
        Bf[kb].y = wmat(targ_w, null_w, beam, fc, k0 + 1, j);
    }

    float phi_r = 0.f, phi_i = 0.f, psd = 0.f;   // state for n = lane (lanes 0-3)
    const float a = ALPHA, oma = 1.0f - ALPHA;

    stage_tile(0, 0);                            // prologue DMA

    for (int tt = 0; tt < 32; ++tt) {
        const int buf = tt & 1;
        const int t0  = tt * 16;

        WAIT_ASYNC();                // this wave's staging done
        __syncthreads();             // everyone's staging visible

        if (tt + 1 < 32) stage_tile(tt + 1, buf ^ 1);   // overlap next tile's DMA

        // ---- A fragments from LDS + power partial ----
        const float* sx = s_x[buf];
        v2f Af[4];
        float ss = 0.f;
#pragma unroll
        for (int kb = 0; kb < 4; ++kb) {
            const int k0 = kb*4 + hi*2;
            const int ri = k0 >> 3, c = k0 & 7;
            const float2 v = *reinterpret_cast<const float2*>(
                &sx[(ri*16 + j)*XROW + wave*8 + c]);     // lanes differ in j -> no conflicts
            Af[kb].x = v.x; Af[kb].y = v.y;
            ss += v.x*v.x + v.y*v.y;
        }
        ss += __shfl_xor(ss, 16, 32);            // combine lower/upper K halves per frame
        if (hi == 0) s_pw[wave][j] = ss * (1.0f / NC);

        // ---- D = A x B, K=16 via 4 chained v_wmma_f32_16x16x4_f32 ----
        v8f D = {};
        D = __builtin_amdgcn_wmma_f32_16x16x4_f32(false, Af[0], false, Bf[0], (short)0, D, false, false);
        D = __builtin_amdgcn_wmma_f32_16x16x4_f32(false, Af[1], false, Bf[1], (short)0, D, false, false);
        D = __builtin_amdgcn_wmma_f32_16x16x4_f32(false, Af[2], false, Bf[2], (short)0, D, false, false);
        D = __builtin_amdgcn_wmma_f32_16x16x4_f32(false, Af[3], false, Bf[3], (short)0, D, false, false);

        // D layout: lane holds column N=j; VGPR r is row M=r (+8 for hi lanes)
        if (j < 10) {
#pragma unroll
            for (int r = 0; r < 8; ++r) s_D[wave][r + hi*8][j] = D[r];
        }
        __syncthreads();

        // ---- sequential recurrence over the tile's frames ----
        const int nvalid = (NT - t0) < 16 ? (NT - t0) : 16;
        if (vf) {
            if (lane < NN) {
                const int n = lane;
                for (int m = 0; m < nvalid; ++m) {
                    const int t = t0 + m;
                    const float tr = s_D[wave][m][0];
                    const float ti = s_D[wave][m][1];
                    const float nr = s_D[wave][m][2 + n];
                    const float ni = s_D[wave][m][6 + n];
                    const float pw = s_pw[wave][m];
                    float d;
                    if (t == 0) {
                        phi_r = oma*tr*nr + ti*ni;
                        phi_i = oma*ti*nr - tr*ni;
                        psd   = oma*pw;
                        float phi = sqrtf(phi_r*phi_r + phi_i*phi_i);
                        d = clip01(phi / (psd + 1e-13f));        // no ratio at t=0
                    } else {
                        phi_r = a*phi_r + oma*(tr*nr + ti*ni);
                        phi_i = a*phi_i + oma*(ti*nr - tr*ni);
                        psd   = a*psd   + oma*pw;
                        float phi = sqrtf(phi_r*phi_r + phi_i*phi_i);
                        float dc  = clip01(phi / (psd + 1e-13f));
                        float rt  = ratio_ws[((size_t)b*NT + t)*NN + n];
                        d = sqrtf(dc * rt);
                    }
                    dcf_out[(((size_t)b*NT + t)*NF + f)*NN + n] = d;
                }
            } else if (lane == 4) {
                for (int m = 0; m < nvalid; ++m)
                    targ_out[(((size_t)b*NT + (t0 + m))*2 + 0)*NF + f] = s_D[wave][m][0];
            } else if (lane == 5) {
                for (int m = 0; m < nvalid; ++m)
                    targ_out[(((size_t)b*NT + (t0 + m))*2 + 1)*NF + f] = s_D[wave][m][1];
            }
        }
        __syncthreads();   // protect s_D/s_pw (and staged buffer rotation)
    }
}

extern "C" void kernel_launch(void* const* d_in, const int* in_sizes, int n_in,
                              void* d_out, int out_size, void* d_ws, size_t ws_size,
                              hipStream_t stream) {
    (void)in_sizes; (void)n_in; (void)out_size; (void)ws_size;
    const float* x       = (const float*)d_in[0];
    const int*   beam_id = (const int*)  d_in[1];
    const float* targ_w  = (const float*)d_in[2];
    const float* null_w  = (const float*)d_in[3];

    float* out      = (float*)d_out;
    float* dcf      = out;                               // (8,500,257,4)
    float* targ     = out + (size_t)NB*NT*NF*NN;         // (8,500,2,257)
    float* ratio_ws = (float*)d_ws;                      // (8,500,4) = 64 KB

    hipLaunchKernelGGL(dcf_stage1_ratio, dim3(NB), dim3(288), 0, stream,
                       x, beam_id, targ_w, null_w, ratio_ws);
    hipLaunchKernelGGL(dcf_stage2_main, dim3(NB, (NF + 7) / 8), dim3(256), 0, stream,
                       x, beam_id, targ_w, null_w, ratio_ws, dcf, targ);
}